// H_SS2D_62045097558411
// MI455X (gfx1250) — compile-verified
//
#include <hip/hip_runtime.h>
#include <math.h>

typedef float v2f __attribute__((ext_vector_type(2)));
typedef float v8f __attribute__((ext_vector_type(8)));

#define LPIX 2304
#define HH 48
#define WW 48

// ---------------------------------------------------------------------------
// Batched WMMA fp32 GEMM:  C[m,n] = sum_k A[m,k] * B[k,n]  (+bias, +softplus)
// One wave computes a 16x64 strip: 4 independent 16x16 accumulators sharing
// the A fragment -> 4 v_wmma_f32_16x16x4_f32 per (2 A + 8 B) lane loads.
// N must be a multiple of 64 (always 2304 here).
// A-row indices are clamped, not guarded: garbage rows only affect D rows
// >= M, which are never stored. K tail zeros the B side branchlessly.
// epi: 0=none, 1=+bias[m], 2=softplus(x+bias[m])
// ---------------------------------------------------------------------------
__global__ void gemm_wmma_f32(
    const float* __restrict__ A, int lda, long sA, int aMod,
    const float* __restrict__ B, int ldb, long sB,
    float* __restrict__ C, int ldc, long sC,
    const float* __restrict__ bias, long sBias, int biasMod,
    int M, int N, int K, int epi)
{
  int batch = blockIdx.z;
  const float* Ab = A + (size_t)(batch % aMod) * sA;
  const float* Bb = B + (size_t)batch * sB;
  float*       Cb = C + (size_t)batch * sC;
  const float* biasb = (epi >= 1) ? (bias + (size_t)(batch % biasMod) * sBias) : nullptr;

  int wid  = threadIdx.x >> 5;
  int lane = threadIdx.x & 31;
  int half = lane >> 4;     // 0 = lanes 0-15, 1 = lanes 16-31
  int l16  = lane & 15;

  int m0 = blockIdx.y * 64 + wid * 16;
  int n0 = blockIdx.x * 64;
  if (m0 >= M) return;      // uniform per wave

  int mA  = m0 + l16;
  int mAc = (mA < M) ? mA : (M - 1);                 // branchless clamp
  const float* Arow = Ab + (size_t)mAc * lda + 2 * half;        // [kk + r]
  const float* Bcol = Bb + n0 + l16 + (size_t)(2 * half) * ldb; // [(kk+r)*ldb + 16t]

  v8f acc[4];
#pragma unroll
  for (int t = 0; t < 4; ++t) acc[t] = (v8f){0.f,0.f,0.f,0.f,0.f,0.f,0.f,0.f};

  int K4 = K & ~3;
#pragma unroll 2
  for (int kk = 0; kk < K4; kk += 4) {
    v2f a, b0, b1, b2, b3;
#pragma unroll
    for (int r = 0; r < 2; ++r) {
      size_t ko = (size_t)(kk + r) * ldb;
      a[r]  = Arow[kk + r];
      b0[r] = Bcol[ko];
      b1[r] = Bcol[ko + 16];
      b2[r] = Bcol[ko + 32];
      b3[r] = Bcol[ko + 48];
    }
    acc[0] = __builtin_amdgcn_wmma_f32_16x16x4_f32(false, a, false, b0, (short)0, acc[0], false, false);
    acc[1] = __builtin_amdgcn_wmma_f32_16x16x4_f32(false, a, false, b1, (short)0, acc[1], false, false);
    acc[2] = __builtin_amdgcn_wmma_f32_16x16x4_f32(false, a, false, b2, (short)0, acc[2], false, false);
    acc[3] = __builtin_amdgcn_wmma_f32_16x16x4_f32(false, a, false, b3, (short)0, acc[3], false, false);
  }
  if (K4 < K) {   // K%4 != 0 tail (only the dt projection, R in {1,2})
    v2f a, b0, b1, b2, b3;
#pragma unroll
    for (int r = 0; r < 2; ++r) {
      int k  = K4 + 2 * half + r;
      int kc = (k < K) ? k : 0;                      // clamped address
      float z = (k < K) ? 1.f : 0.f;                 // branchless zeroing
      size_t ko = (size_t)kc * ldb + n0 + l16;
      a[r]  = Ab[(size_t)mAc * lda + kc];
      b0[r] = Bb[ko]      * z;
      b1[r] = Bb[ko + 16] * z;
      b2[r] = Bb[ko + 32] * z;
      b3[r] = Bb[ko + 48] * z;
    }
    acc[0] = __builtin_amdgcn_wmma_f32_16x16x4_f32(false, a, false, b0, (short)0, acc[0], false, false);
    acc[1] = __builtin_amdgcn_wmma_f32_16x16x4_f32(false, a, false, b1, (short)0, acc[1], false, false);
    acc[2] = __builtin_amdgcn_wmma_f32_16x16x4_f32(false, a, false, b2, (short)0, acc[2], false, false);
    acc[3] = __builtin_amdgcn_wmma_f32_16x16x4_f32(false, a, false, b3, (short)0, acc[3], false, false);
  }

  // C/D layout: VGPR r = row m0+r (lanes 0-15) / m0+8+r (lanes 16-31)
  int n = n0 + l16;
  if (m0 + 16 <= M) {          // uniform: full tile, no guards
#pragma unroll
    for (int t = 0; t < 4; ++t) {
#pragma unroll
      for (int r = 0; r < 8; ++r) {
        int m = m0 + r + 8 * half;
        float v = acc[t][r];
        if (epi >= 1) v += biasb[m];
        if (epi == 2) v = (v > 20.f) ? v : log1pf(__expf(v));
        Cb[(size_t)m * ldc + n + 16 * t] = v;
      }
    }
  } else {                     // M edge tile (x_dbl only; epi==0 there)
#pragma unroll
    for (int t = 0; t < 4; ++t) {
#pragma unroll
      for (int r = 0; r < 8; ++r) {
        int m = m0 + r + 8 * half;
        int mc = (m < M) ? m : (M - 1);
        float v = acc[t][r];
        if (epi >= 1) v += biasb[mc];
        if (epi == 2) v = (v > 20.f) ? v : log1pf(__expf(v));
        if (m < M) Cb[(size_t)m * ldc + n + 16 * t] = v;
      }
    }
  }
}

// ---------------------------------------------------------------------------
// Depthwise conv (rad=1: 3x3, rad=3: 7x7), optional SiLU. NCHW channel-major.
// ---------------------------------------------------------------------------
__global__ void dwconv_k(const float* __restrict__ src, long srcB,
                         float* __restrict__ dst, long dstB,
                         const float* __restrict__ w, const float* __restrict__ bias,
                         int C, int rad, int doSilu, int nb)
{
  size_t idx = (size_t)blockIdx.x * blockDim.x + threadIdx.x;
  size_t total = (size_t)nb * C * LPIX;
  if (idx >= total) return;
  int l = (int)(idx % LPIX);
  int c = (int)((idx / LPIX) % C);
  int b = (int)(idx / ((size_t)C * LPIX));
  int h = l / WW, x = l % WW;
  int ks = 2 * rad + 1;
  const float* wc = w + (size_t)c * ks * ks;
  const float* sc = src + (size_t)b * srcB + (size_t)c * LPIX;
  float acc = bias[c];
  for (int dy = -rad; dy <= rad; ++dy) {
    int yy = h + dy;
    if (yy < 0 || yy >= HH) continue;
    for (int dx = -rad; dx <= rad; ++dx) {
      int xx = x + dx;
      if (xx < 0 || xx >= WW) continue;
      acc += wc[(dy + rad) * ks + (dx + rad)] * sc[yy * WW + xx];
    }
  }
  if (doSilu) acc = acc / (1.f + __expf(-acc));
  dst[(size_t)b * dstB + (size_t)c * LPIX + l] = acc;
}

// out[b,c,l] = in[b,c,l] * dw[b, off+c, l]   (dw has 496 channels)
__global__ void emul_k(const float* __restrict__ in, long inB,
                       const float* __restrict__ dw,
                       float* __restrict__ out, int C, int off, int nb)
{
  size_t idx = (size_t)blockIdx.x * blockDim.x + threadIdx.x;
  size_t total = (size_t)nb * C * LPIX;
  if (idx >= total) return;
  int l = (int)(idx % LPIX);
  int c = (int)((idx / LPIX) % C);
  int b = (int)(idx / ((size_t)C * LPIX));
  out[((size_t)b * C + c) * LPIX + l] =
      in[(size_t)b * inB + (size_t)c * LPIX + l] *
      dw[((size_t)b * 496 + (off + c)) * LPIX + l];
}

// scatter xc (B,di,L) into 4 scan directions xs (B,4,di,L)
__global__ void build_xs_k(const float* __restrict__ xc, float* __restrict__ xs,
                           int di, int nb)
{
  size_t idx = (size_t)blockIdx.x * blockDim.x + threadIdx.x;
  size_t total = (size_t)nb * di * LPIX;
  if (idx >= total) return;
  int l = (int)(idx % LPIX);
  int d = (int)((idx / LPIX) % di);
  int b = (int)(idx / ((size_t)di * LPIX));
  float v = xc[((size_t)b * di + d) * LPIX + l];
  int h = l / WW, w = l % WW;
  int lwh = w * HH + h;
  size_t ds = (size_t)di * LPIX;
  size_t base = (size_t)b * 4 * ds + (size_t)d * LPIX;
  xs[base + 0 * ds + l] = v;
  xs[base + 1 * ds + lwh] = v;
  xs[base + 2 * ds + (LPIX - 1 - l)] = v;
  xs[base + 3 * ds + (LPIX - 1 - lwh)] = v;
}

// ---------------------------------------------------------------------------
// Selective scan. One thread per (b,dir,channel); 16 states in registers.
// delta/u/B/C staged through LDS with coalesced cooperative loads; row
// stride padded to 33 words to avoid 64-bank conflicts on the compute reads.
// ys may alias delta (each element is staged to LDS, then a barrier, before its overwrite).
// ---------------------------------------------------------------------------
#define SCAN_BS 128
#define SCAN_CL 32
#define SCAN_ST 33   // padded row stride
__global__ void scan_k(const float* delta, const float* __restrict__ xs,
                       const float* __restrict__ xdbl,
                       const float* __restrict__ A_log, const float* __restrict__ Dp,
                       float* ys, int di, int R, int Cd)
{
  __shared__ float Dsh[SCAN_BS * SCAN_ST];
  __shared__ float Ush[SCAN_BS * SCAN_ST];
  __shared__ float Bsh[SCAN_CL * 16];
  __shared__ float Csh[SCAN_CL * 16];

  int b = blockIdx.z, k = blockIdx.y;
  int dBase = blockIdx.x * SCAN_BS;
  int d = dBase + threadIdx.x;
  int bk = b * 4 + k;
  bool act = (d < di);
  int nRows = di - dBase; if (nRows > SCAN_BS) nRows = SCAN_BS;

  float Ac[16], h[16];
  float Dv = 0.f;
  if (act) {
#pragma unroll
    for (int n = 0; n < 16; ++n) {
      Ac[n] = -__expf(A_log[((size_t)k * di + d) * 16 + n]);
      h[n] = 0.f;
    }
    Dv = Dp[(size_t)k * di + d];
  }
  const float* dltB = delta + ((size_t)bk * di + dBase) * LPIX;
  const float* uB   = xs    + ((size_t)bk * di + dBase) * LPIX;
  float*       y_   = ys    + ((size_t)bk * di + d) * LPIX;
  const float* xb   = xdbl  + (size_t)bk * Cd * LPIX;

  for (int l0 = 0; l0 < LPIX; l0 += SCAN_CL) {
    __syncthreads();   // previous chunk's compute done before LDS overwrite
    for (int idx = threadIdx.x; idx < nRows * SCAN_CL; idx += SCAN_BS) {
      int dr = idx / SCAN_CL;            // consecutive tid -> consecutive l
      int li = idx % SCAN_CL;
      Dsh[dr * SCAN_ST + li] = dltB[(size_t)dr * LPIX + l0 + li];
      Ush[dr * SCAN_ST + li] = uB  [(size_t)dr * LPIX + l0 + li];
    }
    for (int idx = threadIdx.x; idx < 16 * SCAN_CL; idx += SCAN_BS) {
      int n = idx / SCAN_CL;
      int li = idx % SCAN_CL;
      Bsh[li * 16 + n] = xb[(size_t)(R + n) * LPIX + l0 + li];
      Csh[li * 16 + n] = xb[(size_t)(R + 16 + n) * LPIX + l0 + li];
    }
    __syncthreads();
    if (act) {
      int dr = threadIdx.x;
      for (int li = 0; li < SCAN_CL; ++li) {
        float dl = Dsh[dr * SCAN_ST + li];
        float u  = Ush[dr * SCAN_ST + li];
        float du = dl * u;
        float accv = 0.f;
#pragma unroll
        for (int n = 0; n < 16; ++n) {
          float e = __expf(dl * Ac[n]);
          h[n] = e * h[n] + du * Bsh[li * 16 + n];
          accv += h[n] * Csh[li * 16 + n];
        }
        y_[l0 + li] = accv + Dv * u;     // includes D*xs term
      }
    }
  }
}

// y[b,d,l] = ys0[l] + ys2[L-1-l] + ys1[wh(l)] + ys3[L-1-wh(l)]
__global__ void merge_k(const float* __restrict__ ys, float* __restrict__ y,
                        int di, int nb)
{
  size_t idx = (size_t)blockIdx.x * blockDim.x + threadIdx.x;
  size_t total = (size_t)nb * di * LPIX;
  if (idx >= total) return;
  int l = (int)(idx % LPIX);
  int d = (int)((idx / LPIX) % di);
  int b = (int)(idx / ((size_t)di * LPIX));
  int h = l / WW, w = l % WW;
  int lwh = w * HH + h;
  size_t ds = (size_t)di * LPIX;
  size_t base = (size_t)b * 4 * ds + (size_t)d * LPIX;
  y[((size_t)b * di + d) * LPIX + l] =
      ys[base + l] +
      ys[base + 2 * ds + (LPIX - 1 - l)] +
      ys[base + 1 * ds + lwh] +
      ys[base + 3 * ds + (LPIX - 1 - lwh)];
}

// LayerNorm over channels + SiLU(z) gate. One wave32 per pixel.
__global__ void ln_gate_k(float* __restrict__ y, const float* __restrict__ xz,
                          const float* __restrict__ lnw, const float* __restrict__ lnb,
                          int di, int dm4, int nb)
{
  int wid = threadIdx.x >> 5, lane = threadIdx.x & 31;
  int p = blockIdx.x * (blockDim.x >> 5) + wid;
  if (p >= nb * LPIX) return;
  int b = p / LPIX, l = p % LPIX;
  float* yr = y + (size_t)b * di * LPIX + l;
  const float* zr = xz + ((size_t)b * dm4 + di) * LPIX + l;
  float s = 0.f, ss = 0.f;
  for (int d = lane; d < di; d += 32) {
    float v = yr[(size_t)d * LPIX];
    s += v; ss += v * v;
  }
#pragma unroll
  for (int m = 16; m >= 1; m >>= 1) {
    s  += __shfl_xor(s,  m, 32);
    ss += __shfl_xor(ss, m, 32);
  }
  float mu = s / di;
  float var = ss / di - mu * mu;
  float rstd = rsqrtf(var + 1e-5f);
  for (int d = lane; d < di; d += 32) {
    float v = (yr[(size_t)d * LPIX] - mu) * rstd * lnw[d] + lnb[d];
    float z = zr[(size_t)d * LPIX];
    yr[(size_t)d * LPIX] = v * (z / (1.f + __expf(-z)));
  }
}

// ---------------------------------------------------------------------------
// Host-side orchestration
// ---------------------------------------------------------------------------
static inline dim3 grid1d(size_t total, int bs) {
  return dim3((unsigned)((total + bs - 1) / bs));
}

struct WS { float *dw, *h, *xz, *xc, *xs, *xdbl, *dlt; };

static void run_gemm(hipStream_t st,
                     const float* A, int lda, long sA, int aMod,
                     const float* B, int ldb, long sB,
                     float* C, int ldc, long sC,
                     const float* bias, long sBias, int biasMod,
                     int M, int N, int K, int epi, int batch)
{
  dim3 g((N + 63) / 64, (M + 63) / 64, batch);
  gemm_wmma_f32<<<g, dim3(128), 0, st>>>(A, lda, sA, aMod, B, ldb, sB,
                                         C, ldc, sC, bias, sBias, biasMod,
                                         M, N, K, epi);
}

// P: in_w,in_b,conv_w,conv_b,xproj_w,dt_w,dt_b,A_log,D,ln_w,ln_b,out_w,out_b
static void run_ss2d(hipStream_t st, const float* const* P, int dm, const WS& w)
{
  int di = 2 * dm;
  int R = (dm >= 16) ? dm / 16 : 1;
  int Cd = R + 32;
  int dm4 = 4 * dm;

  // xz = in_w @ h + in_b : (B, 4dm, L)
  run_gemm(st, P[0], dm, 0, 1, w.h, LPIX, (long)dm * LPIX,
           w.xz, LPIX, (long)dm4 * LPIX, P[1], 0, 1, dm4, LPIX, dm, 1, 2);
  // xc = silu(dwconv3(xz rows [0,di)))
  dwconv_k<<<grid1d(2ull * di * LPIX, 256), 256, 0, st>>>(
      w.xz, (long)dm4 * LPIX, w.xc, (long)di * LPIX, P[2], P[3], di, 1, 1, 2);
  // 4 scan directions
  build_xs_k<<<grid1d(2ull * di * LPIX, 256), 256, 0, st>>>(w.xc, w.xs, di, 2);
  // x_dbl = xproj_w[k] @ xs[b,k] : (B,4,Cd,L)
  run_gemm(st, P[4], di, (long)Cd * di, 4, w.xs, LPIX, (long)di * LPIX,
           w.xdbl, LPIX, (long)Cd * LPIX, nullptr, 0, 1, Cd, LPIX, di, 0, 8);
  // delta = softplus(dt_w[k] @ dts + dt_b[k]) : (B,4,di,L)
  run_gemm(st, P[5], R, (long)di * R, 4, w.xdbl, LPIX, (long)Cd * LPIX,
           w.dlt, LPIX, (long)di * LPIX, P[6], di, 4, di, LPIX, R, 2, 8);
  // selective scan; ys written in-place over delta
  scan_k<<<dim3((di + SCAN_BS - 1) / SCAN_BS, 4, 2), SCAN_BS, 0, st>>>(
      w.dlt, w.xs, w.xdbl, P[7], P[8], w.dlt, di, R, Cd);
  // merge 4 directions into xc (reused as y)
  merge_k<<<grid1d(2ull * di * LPIX, 256), 256, 0, st>>>(w.dlt, w.xc, di, 2);
  // layernorm + silu(z) gate, in-place on xc
  ln_gate_k<<<grid1d((size_t)2 * LPIX * 32, 256), 256, 0, st>>>(
      w.xc, w.xz, P[9], P[10], di, dm4, 2);
  // h = out_w @ y + out_b : (B, dm, L)
  run_gemm(st, P[11], di, 0, 1, w.xc, LPIX, (long)di * LPIX,
           w.h, LPIX, (long)dm * LPIX, P[12], 0, 1, dm, LPIX, di, 1, 2);
}

extern "C" void kernel_launch(void* const* d_in, const int* in_sizes, int n_in,
                              void* d_out, int out_size, void* d_ws, size_t ws_size,
                              hipStream_t stream)
{
  (void)in_sizes; (void)n_in; (void)out_size; (void)ws_size;
  const float* const* In = (const float* const*)d_in;
  const float* x   = In[0];
  const float* piw = In[1];
  const float* pib = In[2];
  const float* dww = In[3];
  const float* dwb = In[4];

  float* ws = (float*)d_ws;
  WS w;
  size_t o = 0;
  w.dw   = ws + o; o += 2ull * 496 * LPIX;       // depthwise output (496 ch)
  w.h    = ws + o; o += 2ull * 256 * LPIX;       // running activation h
  w.xz   = ws + o; o += 2ull * 1024 * LPIX;      // in-proj output (xc|z)
  w.xc   = ws + o; o += 2ull * 512 * LPIX;       // conv/silu out, later y
  w.xs   = ws + o; o += 2ull * 4 * 512 * LPIX;   // 4-direction inputs
  w.xdbl = ws + o; o += 2ull * 4 * 48 * LPIX;    // dts|B|C
  w.dlt  = ws + o; o += 2ull * 4 * 512 * LPIX;   // delta, then ys (in-place)
  float* fused = w.xs;   // alias: fused is dead before xs is first written

  // fused = proj_in_w @ x + b : (B,512,L)
  run_gemm(stream, piw, 256, 0, 1, x, LPIX, 256L * LPIX,
           fused, LPIX, 512L * LPIX, pib, 0, 1, 512, LPIX, 256, 1, 2);
  // dw_abc = dwconv7(fused rows [16,512))
  dwconv_k<<<grid1d(2ull * 496 * LPIX, 256), 256, 0, stream>>>(
      fused + 16 * LPIX, 512L * LPIX, w.dw, 496L * LPIX, dww, dwb, 496, 3, 0, 2);
  // h0 = pwa * dw_list[0]
  emul_k<<<grid1d(2ull * 16 * LPIX, 256), 256, 0, stream>>>(
      fused, 512L * LPIX, w.dw, w.h, 16, 0, 2);

  static const int DIMS[5] = {16, 32, 64, 128, 256};
  static const int CUM[5]  = {0, 16, 48, 112, 240};  // dw_list row offsets

  const float* P0[13];
  for (int j = 0; j < 13; ++j) P0[j] = In[15 + j];
  run_ss2d(stream, P0, DIMS[0], w);

  for (int i = 0; i < 4; ++i) {
    int Di = DIMS[i], Do = DIMS[i + 1];
    // tmp = pw_ws[i] @ h + pw_bs[i]
    run_gemm(stream, In[5 + i], Di, 0, 1, w.h, LPIX, (long)Di * LPIX,
             w.xc, LPIX, (long)Do * LPIX, In[9 + i], 0, 1, Do, LPIX, Di, 1, 2);
    // h = tmp * dw_list[i+1]
    emul_k<<<grid1d(2ull * Do * LPIX, 256), 256, 0, stream>>>(
        w.xc, (long)Do * LPIX, w.dw, w.h, Do, CUM[i + 1], 2);
    const float* Pp[13];
    for (int j = 0; j < 13; ++j) Pp[j] = In[15 + 13 * (i + 1) + j];
    run_ss2d(stream, Pp, Do, w);
  }

  // out = proj_out_w @ h + proj_out_b : (B,256,L)
  run_gemm(stream, In[13], 256, 0, 1, w.h, LPIX, 256L * LPIX,
           (float*)d_out, LPIX, 256L * LPIX, In[14], 0, 1, 256, LPIX, 256, 1, 2);
}